// UniversityDistributionLoss_38474317038467
// MI455X (gfx1250) — compile-verified
//
#include <hip/hip_runtime.h>
#include <hip/hip_bf16.h>

typedef float v2f __attribute__((ext_vector_type(2)));
typedef float v8f __attribute__((ext_vector_type(8)));

#define B_ROWS 256
#define D_COLS 512
#define KTOT   1024   // concatenated K: [D | D]

// ---------------------------------------------------------------------------
// Kernel 1: per-row transform + row reductions.
//   Ap[r, 0:512]   = mu^2 + exp(lv)         Ap[r, 512:1024] = mu
//   Bp[r, 0:512]   = exp(-lv)               Bp[r, 512:1024] = -2*mu*exp(-lv)
//   Slv[r] = sum_d lv[r,d]     Cr[r] = sum_d mu^2 * exp(-lv)
// One block per row, 256 threads, 2 columns each, LDS tree reduce.
// ---------------------------------------------------------------------------
__global__ void kl_prep_kernel(const float* __restrict__ mu,
                               const float* __restrict__ lv,
                               float* __restrict__ Ap,
                               float* __restrict__ Bp,
                               float* __restrict__ Slv,
                               float* __restrict__ Cr) {
    __shared__ float s1[256];
    __shared__ float s2[256];
    const int r  = blockIdx.x;
    const int td = threadIdx.x;

    float slv_acc = 0.0f;
    float c_acc   = 0.0f;
    #pragma unroll
    for (int d = td; d < D_COLS; d += 256) {
        const float m = mu[r * D_COLS + d];
        const float l = lv[r * D_COLS + d];
        const float var  = expf(l);
        const float ivar = expf(-l);
        Ap[r * KTOT + d]          = m * m + var;
        Ap[r * KTOT + D_COLS + d] = m;
        Bp[r * KTOT + d]          = ivar;
        Bp[r * KTOT + D_COLS + d] = -2.0f * m * ivar;
        slv_acc += l;
        c_acc   += m * m * ivar;
    }
    s1[td] = slv_acc;
    s2[td] = c_acc;
    __syncthreads();
    for (int s = 128; s > 0; s >>= 1) {
        if (td < s) { s1[td] += s1[td + s]; s2[td] += s2[td + s]; }
        __syncthreads();
    }
    if (td == 0) { Slv[r] = s1[0]; Cr[r] = s2[0]; }
}

// ---------------------------------------------------------------------------
// Kernel 2: fused GEMM M = A' @ B'^T via V_WMMA_F32_16X16X4_F32 + epilogue.
// 8 waves per block, one 16x16 output tile per wave, 256 tiles total.
// f32 16x4 A layout (ISA 7.12.2): lanes 0-15 hold rows M=0..15 with K={k,k+1},
// lanes 16-31 hold K={k+2,k+3}. B (4x16) is symmetric with N on lanes, so for
// B'^T both sides use identical per-lane row-major addressing.
// ---------------------------------------------------------------------------
__global__ void kl_wmma_kernel(const float* __restrict__ Ap,
                               const float* __restrict__ Bp,
                               const float* __restrict__ Slv,
                               const float* __restrict__ Cr,
                               float* __restrict__ partial) {
    const int wave = threadIdx.x >> 5;          // 0..7
    const int lane = threadIdx.x & 31;
    const int t    = blockIdx.x * 8 + wave;     // tile id 0..255
    const int ti   = t >> 4;                    // row-tile
    const int tj   = t & 15;                    // col-tile
    const int half = lane >> 4;                 // 0: K={k,k+1}  1: K={k+2,k+3}
    const int lrow = lane & 15;                 // matrix row (A) / col (B)

    const v2f* __restrict__ a2 =
        (const v2f*)(Ap + (size_t)(ti * 16 + lrow) * KTOT + half * 2);
    const v2f* __restrict__ b2 =
        (const v2f*)(Bp + (size_t)(tj * 16 + lrow) * KTOT + half * 2);

    v8f c = {};
    #pragma unroll 8
    for (int k = 0; k < KTOT; k += 4) {
        v2f a = a2[k >> 1];                     // global_load_b64
        v2f b = b2[k >> 1];                     // global_load_b64
        // 8 args: (neg_a, A, neg_b, B, c_mod, C, reuse_a, reuse_b)
        c = __builtin_amdgcn_wmma_f32_16x16x4_f32(
                false, a, false, b, (short)0, c, false, false);
    }

    // Epilogue: D layout — lane<16: c[v] = M[ti*16+v][tj*16+lane]
    //                      lane>=16: c[v] = M[ti*16+8+v][tj*16+lane-16]
    const int j    = tj * 16 + lrow;
    const float sj = Slv[j];
    const float cj = Cr[j];
    const int ibase = ti * 16 + half * 8;

    float acc = 0.0f;
    #pragma unroll
    for (int v = 0; v < 8; ++v) {
        const int i = ibase + v;
        const float e = c[v] + sj - Slv[i] + cj - (float)D_COLS;
        acc += (i == j) ? 0.0f : e;             // diagonal mask (1 - eye)
    }

    // wave32 butterfly reduction
    #pragma unroll
    for (int o = 16; o > 0; o >>= 1) acc += __shfl_xor(acc, o, 32);
    if (lane == 0) partial[t] = acc;
}

// ---------------------------------------------------------------------------
// Kernel 3: deterministic 256 -> 1 tree reduction, scale by 0.5 / B.
// ---------------------------------------------------------------------------
__global__ void kl_finish_kernel(const float* __restrict__ partial,
                                 float* __restrict__ out) {
    __shared__ float s[256];
    const int td = threadIdx.x;
    s[td] = partial[td];
    __syncthreads();
    for (int sz = 128; sz > 0; sz >>= 1) {
        if (td < sz) s[td] += s[td + sz];
        __syncthreads();
    }
    if (td == 0) out[0] = s[0] * (0.5f / (float)B_ROWS);
}

// ---------------------------------------------------------------------------
extern "C" void kernel_launch(void* const* d_in, const int* in_sizes, int n_in,
                              void* d_out, int out_size, void* d_ws, size_t ws_size,
                              hipStream_t stream) {
    (void)in_sizes; (void)n_in; (void)out_size; (void)ws_size;
    const float* mu = (const float*)d_in[0];
    const float* lv = (const float*)d_in[1];
    float* out      = (float*)d_out;

    // Workspace layout (bytes):
    //   Ap:      [0,            1 MB)   256 x 1024 f32
    //   Bp:      [1 MB,         2 MB)   256 x 1024 f32
    //   Slv:     [2 MB,         +1 KB)  256 f32
    //   Cr:      [+1 KB,        +2 KB)  256 f32
    //   partial: [+2 KB,        +3 KB)  256 f32
    char* ws = (char*)d_ws;
    float* Ap      = (float*)(ws);
    float* Bp      = (float*)(ws + (size_t)B_ROWS * KTOT * sizeof(float));
    float* Slv     = (float*)(ws + (size_t)2 * B_ROWS * KTOT * sizeof(float));
    float* Cr      = Slv + B_ROWS;
    float* partial = Cr + B_ROWS;

    kl_prep_kernel<<<B_ROWS, 256, 0, stream>>>(mu, lv, Ap, Bp, Slv, Cr);
    kl_wmma_kernel<<<32, 256, 0, stream>>>(Ap, Bp, Slv, Cr, partial);
    kl_finish_kernel<<<1, 256, 0, stream>>>(partial, out);
}